// MS_SSIM_1D_Loss_38792144617548
// MI455X (gfx1250) — compile-verified
//
#include <hip/hip_runtime.h>
#include <hip/hip_bf16.h>

typedef __attribute__((ext_vector_type(2))) float v2f;
typedef __attribute__((ext_vector_type(8))) float v8f;
typedef int v4i_ld __attribute__((vector_size(16)));   // matches builtin param type

#define NTHREADS 256
#define NWAVES   8          // 256 / wave32
#define LEVELS   5
#define BATCH    32
#define CHAN     64
#define L0       16384
#define C1_CONST 1.0e-4f    // (K1*DATA_RANGE)^2
#define C2_CONST 9.0e-4f    // (K2*DATA_RANGE)^2

#if defined(__HIP_DEVICE_COMPILE__) && __has_builtin(__builtin_amdgcn_global_load_async_to_lds_b128)
#define HAVE_ASYNC_LDS 1
#else
#define HAVE_ASYNC_LDS 0
#endif

// Normalized 11-tap Gaussian, sigma = 1.5 (matches reference _gauss_window)
__constant__ float GW[11] = {
    0.00102838f, 0.00759876f, 0.03600077f, 0.10936069f, 0.21300553f,
    0.26601172f, 0.21300553f, 0.10936069f, 0.03600077f, 0.00759876f,
    0.00102838f};

// workspace (floats): [0..159] cs sums (lvl*32+b), [160..319] ssim sums,
// [320..351] grad sums per batch, [352] global l1 sum
#define ACC_CS    0
#define ACC_SSIM  160
#define ACC_GRAD  320
#define ACC_L1    352
#define ACC_TOTAL 353

__global__ __launch_bounds__(512) void msssim_init_kernel(float* acc) {
    int t = threadIdx.x;
    if (t < ACC_TOTAL) acc[t] = 0.0f;
}

__device__ __forceinline__ float lds_or_zero(const float* buf, int i, int n) {
    // zero 'same' padding at row boundaries
    return (i >= 0 && i < n) ? buf[i] : 0.0f;
}

// Reduce two values at once: one barrier sequence instead of two.
__device__ __forceinline__ void block_reduce2(float& a, float& b, float* buf) {
    const int t = threadIdx.x;
    __syncthreads();
    buf[t] = a;
    buf[NTHREADS + t] = b;
    __syncthreads();
#pragma unroll
    for (int s = NTHREADS / 2; s > 0; s >>= 1) {
        if (t < s) {
            buf[t] += buf[t + s];
            buf[NTHREADS + t] += buf[NTHREADS + t + s];
        }
        __syncthreads();
    }
    a = buf[0];
    b = buf[NTHREADS];
    __syncthreads();
}

// One workgroup per (b,c) row. Full 5-level pyramid fused in LDS; conv runs on
// the matrix pipe via V_WMMA_F32_16X16X4_F32 (16 outputs x 16 segments per tile,
// banded-weight A matrix, K = 26 -> padded 28 -> 7 chained K=4 WMMAs).
__global__ __launch_bounds__(NTHREADS) void msssim_row_kernel(
    const float* __restrict__ x, const float* __restrict__ y,
    float* __restrict__ acc) {
    extern __shared__ float smem[];
    float* sx = smem;            // 16384 f
    float* sy = smem + L0;       // 16384 f
    float* sp = smem + 2 * L0;   // 8192 f scratch (pooling + reductions)

    const int t    = threadIdx.x;
    const int row  = blockIdx.x;          // b*CHAN + c
    const int b    = row >> 6;            // row / CHAN
    const int lane = t & 31;
    const int wid  = t >> 5;
    const int m    = lane & 15;           // A: row M / B: column N (ISA f32 layout)
    const int kh   = lane >> 4;           // K half-select (lanes 16..31 -> K+2)

    // ---- load full row once; async global->LDS (ASYNCcnt, no VGPR staging) ----
    {
        const float4* xr = (const float4*)(x + (size_t)row * L0);
        const float4* yr = (const float4*)(y + (size_t)row * L0);
        float4* sx4 = (float4*)sx;
        float4* sy4 = (float4*)sy;
#if HAVE_ASYNC_LDS
        for (int i = t; i < L0 / 4; i += NTHREADS) {
            __builtin_amdgcn_global_load_async_to_lds_b128(
                (__attribute__((address_space(1))) v4i_ld*)(xr + i),
                (__attribute__((address_space(3))) v4i_ld*)(sx4 + i), 0, 0);
            __builtin_amdgcn_global_load_async_to_lds_b128(
                (__attribute__((address_space(1))) v4i_ld*)(yr + i),
                (__attribute__((address_space(3))) v4i_ld*)(sy4 + i), 0, 0);
        }
#if __has_builtin(__builtin_amdgcn_s_wait_asynccnt)
        __builtin_amdgcn_s_wait_asynccnt(0);
#else
        asm volatile("s_wait_asynccnt 0x0" ::: "memory");
#endif
#else
        for (int i = t; i < L0 / 4; i += NTHREADS) {
            sx4[i] = xr[i];
            sy4[i] = yr[i];
        }
#endif
    }
    __syncthreads();

    // ---- banded Gaussian A matrices: W[p,k] = gw[k-p], K in [0,28) ----
    v2f A[7];
#pragma unroll
    for (int j = 0; j < 7; ++j) {
        int k0 = 4 * j + 2 * kh;
        int d0 = k0 - m, d1 = k0 + 1 - m;
        int c0 = min(max(d0, 0), 10), c1 = min(max(d1, 0), 10);
        A[j].x = (d0 >= 0 && d0 < 11) ? GW[c0] : 0.0f;
        A[j].y = (d1 >= 0 && d1 < 11) ? GW[c1] : 0.0f;
    }

    int N = L0;
#pragma unroll 1
    for (int lvl = 0; lvl < LEVELS; ++lvl) {
        const int ntiles = N >> 8;   // 256 outputs per WMMA tile
        float cs_part = 0.0f, ss_part = 0.0f;

        for (int tile = wid; tile < ntiles; tile += NWAVES) {  // wave-uniform
            const int base = tile << 8;
            v8f cx = {}, cy = {}, cxx = {}, cyy = {}, cxy = {};
#pragma unroll
            for (int j = 0; j < 7; ++j) {
                // B[k,n] = in[base + 16n + k - 5]; lane covers (n=m, K=2*kh+{0,1})
                const int i0 = base + m * 16 + 4 * j + 2 * kh - 5;
                const float bx0 = lds_or_zero(sx, i0, N);
                const float bx1 = lds_or_zero(sx, i0 + 1, N);
                const float by0 = lds_or_zero(sy, i0, N);
                const float by1 = lds_or_zero(sy, i0 + 1, N);
                v2f bx  = {bx0, bx1};
                v2f by  = {by0, by1};
                v2f bxx = {bx0 * bx0, bx1 * bx1};
                v2f byy = {by0 * by0, by1 * by1};
                v2f bxy = {bx0 * by0, bx1 * by1};
                cx  = __builtin_amdgcn_wmma_f32_16x16x4_f32(false, A[j], false, bx,  (short)0, cx,  false, false);
                cy  = __builtin_amdgcn_wmma_f32_16x16x4_f32(false, A[j], false, by,  (short)0, cy,  false, false);
                cxx = __builtin_amdgcn_wmma_f32_16x16x4_f32(false, A[j], false, bxx, (short)0, cxx, false, false);
                cyy = __builtin_amdgcn_wmma_f32_16x16x4_f32(false, A[j], false, byy, (short)0, cyy, false, false);
                cxy = __builtin_amdgcn_wmma_f32_16x16x4_f32(false, A[j], false, bxy, (short)0, cxy, false, false);
            }
            // SSIM / CS maps, elementwise (identical accumulator layouts)
#pragma unroll
            for (int r = 0; r < 8; ++r) {
                float mux = cx[r], muy = cy[r];
                float mux2 = mux * mux, muy2 = muy * muy, muxy = mux * muy;
                float sxx = cxx[r] - mux2;
                float syy = cyy[r] - muy2;
                float sxy = cxy[r] - muxy;
                float dcs = sxx + syy + C2_CONST;
                float csn = 2.0f * sxy + C2_CONST;
                float cs  = csn / dcs;
                float ss  = (2.0f * muxy + C1_CONST) * csn /
                            ((mux2 + muy2 + C1_CONST) * dcs);
                cs_part += cs;
                ss_part += ss;
            }
        }

        block_reduce2(cs_part, ss_part, sp);
        if (t == 0) {
            atomicAdd(&acc[ACC_CS + lvl * BATCH + b], cs_part);
            atomicAdd(&acc[ACC_SSIM + lvl * BATCH + b], ss_part);
        }

        if (lvl < LEVELS - 1) {  // avg-pool by 2 (in LDS, via scratch)
            const int half = N >> 1;
            for (int l = t; l < half; l += NTHREADS)
                sp[l] = 0.5f * (sx[2 * l] + sx[2 * l + 1]);
            __syncthreads();
            for (int l = t; l < half; l += NTHREADS) sx[l] = sp[l];
            __syncthreads();
            for (int l = t; l < half; l += NTHREADS)
                sp[l] = 0.5f * (sy[2 * l] + sy[2 * l + 1]);
            __syncthreads();
            for (int l = t; l < half; l += NTHREADS) sy[l] = sp[l];
            __syncthreads();
            N = half;
        }
    }

    // ---- grad + L1 on coarsest level (N == 1024), straight from LDS ----
    float l1s = 0.0f, gds = 0.0f;
    for (int l = t; l < N; l += NTHREADS) {
        l1s += fabsf(sx[l] - sy[l]);
        if (l < N - 1)
            gds += fabsf((sx[l + 1] - sx[l]) - (sy[l + 1] - sy[l]));
    }
    block_reduce2(gds, l1s, sp);
    if (t == 0) {
        atomicAdd(&acc[ACC_GRAD + b], gds);
        atomicAdd(&acc[ACC_L1], l1s);
    }
}

__global__ __launch_bounds__(32) void msssim_final_kernel(
    const float* __restrict__ acc, float* __restrict__ out) {
    const float W[5] = {0.0448f, 0.2856f, 0.3001f, 0.2363f, 0.1333f};
    int bb = threadIdx.x;
    if (bb >= BATCH) return;
    float ms = 1.0f;
    int Nl = L0;
#pragma unroll
    for (int i = 0; i < LEVELS; ++i) {
        float denom = (float)CHAN * (float)Nl;
        float v = (i == LEVELS - 1) ? acc[ACC_SSIM + i * BATCH + bb] / denom
                                    : acc[ACC_CS + i * BATCH + bb] / denom;
        ms *= powf(v, W[i]);
        if (i < LEVELS - 1) Nl >>= 1;
    }
    float grad = acc[ACC_GRAD + bb] / ((float)CHAN * (float)(Nl - 1));
    float l1   = acc[ACC_L1] / ((float)BATCH * (float)CHAN * (float)Nl);
    out[bb] = (1.0f - ms) * 0.5f + l1 * 0.3f + grad * 0.2f;
}

extern "C" void kernel_launch(void* const* d_in, const int* in_sizes, int n_in,
                              void* d_out, int out_size, void* d_ws,
                              size_t ws_size, hipStream_t stream) {
    (void)in_sizes; (void)n_in; (void)out_size; (void)ws_size;
    const float* x = (const float*)d_in[0];
    const float* y = (const float*)d_in[1];
    float* out = (float*)d_out;
    float* acc = (float*)d_ws;

    msssim_init_kernel<<<1, 512, 0, stream>>>(acc);
    const size_t lds_bytes = (size_t)(2 * L0 + L0 / 2) * sizeof(float); // 160 KiB
    msssim_row_kernel<<<BATCH * CHAN, NTHREADS, lds_bytes, stream>>>(x, y, acc);
    msssim_final_kernel<<<1, 32, 0, stream>>>(acc, out);
}